// DomainLoss_85899346229
// MI455X (gfx1250) — compile-verified
//
#include <hip/hip_runtime.h>
#include <math.h>

typedef __attribute__((ext_vector_type(2))) float v2f;
typedef __attribute__((ext_vector_type(8))) float v8f;

#define B_ROWS 2048
#define DIMS   512
#define NTOT   4096
#define BT     128      // block tile (M == N)
#define KT     32       // K panel staged in LDS
#define LDP    36       // padded LDS row stride (mult of 4 for b128, conflict-free banks)
#define NPAN   (DIMS / KT)

__device__ __forceinline__ const float* row_ptr(const float* s, const float* t, int r) {
    return (r < B_ROWS) ? (s + r * DIMS) : (t + (r - B_ROWS) * DIMS);
}

// CDNA5 async copy: global -> LDS without touching VGPRs; tracked by ASYNCcnt.
__device__ __forceinline__ void async_copy_b128(unsigned int lds_byte_off, const float* gptr) {
    asm volatile("global_load_async_to_lds_b128 %0, %1, off"
                 :: "v"(lds_byte_off), "v"(gptr)
                 : "memory");
}
__device__ __forceinline__ void wait_async_8()  { asm volatile("s_wait_asynccnt 0x8" ::: "memory"); }
__device__ __forceinline__ void wait_async_0()  { asm volatile("s_wait_asynccnt 0x0" ::: "memory"); }

__device__ __forceinline__ unsigned int lds_off(const void* p) {
    // generic pointer to LDS: addr[31:0] is the LDS byte offset (CDNA5 aperture rule)
    return (unsigned int)(unsigned long long)p;
}

__global__ void init_kernel(float* accums) {
    if (threadIdx.x < 16) accums[threadIdx.x] = 0.0f;
}

// One block (128 threads) per row: sq[r] = sum_k total[r][k]^2
__global__ void sq_kernel(const float* __restrict__ src, const float* __restrict__ tgt,
                          float* __restrict__ sq) {
    const int row = blockIdx.x;
    const float* p = row_ptr(src, tgt, row);
    float4 v = ((const float4*)p)[threadIdx.x];          // 128 threads * 4 floats = 512
    float s = v.x * v.x + v.y * v.y + v.z * v.z + v.w * v.w;
    for (int off = 16; off; off >>= 1) s += __shfl_down(s, off, 32);
    __shared__ float red[4];
    const int lane = threadIdx.x & 31, wave = threadIdx.x >> 5;
    if (lane == 0) red[wave] = s;
    __syncthreads();
    if (threadIdx.x == 0) sq[row] = red[0] + red[1] + red[2] + red[3];
}

// Single block; writes BCE *sum* into accums[2]
__global__ void bce_kernel(const float* __restrict__ logits, const float* __restrict__ labels,
                           float* __restrict__ accums) {
    float s = 0.0f;
    for (int i = threadIdx.x; i < 2 * B_ROWS; i += blockDim.x) {
        float x = logits[i], y = labels[i];
        s += fmaxf(x, 0.0f) - x * y + log1pf(__expf(-fabsf(x)));
    }
    for (int off = 16; off; off >>= 1) s += __shfl_down(s, off, 32);
    __shared__ float red[8];
    const int lane = threadIdx.x & 31, wave = threadIdx.x >> 5;
    if (lane == 0) red[wave] = s;
    __syncthreads();
    if (threadIdx.x == 0) {
        float t = 0.0f;
        for (int w = 0; w < 8; ++w) t += red[w];
        accums[2] = t;
    }
}

// PHASE 0: accumulate sum of L2 into accums[0]
// PHASE 1: accumulate signed 5-kernel Gaussian sum into accums[1], reading base_bw from accums[0]
template <int PHASE>
__global__ void __launch_bounds__(256)
mmd_gemm(const float* __restrict__ src, const float* __restrict__ tgt,
         const float* __restrict__ sq, float* __restrict__ accums) {
    __shared__ float As[2][BT * LDP];   // double-buffered A panel
    __shared__ float Bs[2][BT * LDP];   // double-buffered B panel
    __shared__ float sqR[BT];
    __shared__ float sqC[BT];
    __shared__ float red[8];

    const int rowBase = blockIdx.x * BT;
    const int colBase = blockIdx.y * BT;
    const int tid   = threadIdx.x;
    const int lane  = tid & 31;
    const int wave  = tid >> 5;
    const int wm    = wave >> 2;      // 0..1 -> 64-row strip
    const int wn    = wave & 3;       // 0..3 -> 32-col strip
    const int lhalf = lane >> 4;      // 0/1 -> K pair {0,1} vs {2,3}
    const int l15   = lane & 15;

    if (tid < BT)       sqR[tid]      = sq[rowBase + tid];
    else                sqC[tid - BT] = sq[colBase + tid - BT];

    // this thread's staging slot: f = tid + u*256 -> (row, col4) inside the 128x32 panel
    const unsigned int asBase = lds_off(&As[0][0]);
    const unsigned int bsBase = lds_off(&Bs[0][0]);
    const unsigned int panBytes = BT * LDP * sizeof(float);

    const v8f vzero = {0.f, 0.f, 0.f, 0.f, 0.f, 0.f, 0.f, 0.f};
    v8f acc[4][2];
#pragma unroll
    for (int mi = 0; mi < 4; ++mi)
#pragma unroll
        for (int ni = 0; ni < 2; ++ni) acc[mi][ni] = vzero;

    // issue panel 0 into buffer 0 (8 async b128 copies per thread: 4 for A, 4 for B)
#pragma unroll
    for (int u = 0; u < 4; ++u) {
        const int f  = tid + u * 256;
        const int r  = f >> 3;
        const int c4 = (f & 7) << 2;
        const unsigned int lo = (unsigned int)((r * LDP + c4) * sizeof(float));
        async_copy_b128(asBase + lo, row_ptr(src, tgt, rowBase + r) + c4);
        async_copy_b128(bsBase + lo, row_ptr(src, tgt, colBase + r) + c4);
    }

    for (int p = 0; p < NPAN; ++p) {
        // prefetch next panel into the other buffer (safe: end-of-compute barrier for
        // panel p-1 already passed, so no wave still reads buffer p^1)
        if (p + 1 < NPAN) {
            const int k0n = (p + 1) * KT;
            const unsigned int bufOff = (unsigned int)((p + 1) & 1) * panBytes;
#pragma unroll
            for (int u = 0; u < 4; ++u) {
                const int f  = tid + u * 256;
                const int r  = f >> 3;
                const int c4 = (f & 7) << 2;
                const unsigned int lo = bufOff + (unsigned int)((r * LDP + c4) * sizeof(float));
                async_copy_b128(asBase + lo, row_ptr(src, tgt, rowBase + r) + k0n + c4);
                async_copy_b128(bsBase + lo, row_ptr(src, tgt, colBase + r) + k0n + c4);
            }
            wait_async_8();   // oldest 8 (panel p) complete; panel p+1 still in flight
        } else {
            wait_async_0();   // last panel complete
        }
        __syncthreads();       // publish panel p LDS data to all waves

        const float* Ab = As[p & 1];
        const float* Bb = Bs[p & 1];
#pragma unroll
        for (int k4 = 0; k4 < KT / 4; ++k4) {
            const int kk = k4 * 4 + lhalf * 2;
            v2f a[4], b[2];
#pragma unroll
            for (int mi = 0; mi < 4; ++mi)
                a[mi] = *(const v2f*)&Ab[(wm * 64 + mi * 16 + l15) * LDP + kk];
#pragma unroll
            for (int ni = 0; ni < 2; ++ni)
                b[ni] = *(const v2f*)&Bb[(wn * 32 + ni * 16 + l15) * LDP + kk];
#pragma unroll
            for (int mi = 0; mi < 4; ++mi)
#pragma unroll
                for (int ni = 0; ni < 2; ++ni)
                    acc[mi][ni] = __builtin_amdgcn_wmma_f32_16x16x4_f32(
                        false, a[mi], false, b[ni], (short)0, acc[mi][ni], false, false);
        }
        __syncthreads();       // all waves done reading buf p&1; next iter may overwrite it
    }

    // Fused epilogue: L2 = max(sq_i + sq_j - 2G, 0); phase-dependent reduction.
    float c0 = 0.f, c1 = 0.f, c2 = 0.f, c3 = 0.f, c4 = 0.f;
    if (PHASE == 1) {
        const float bw  = accums[0] * (1.0f / ((float)NTOT * (float)NTOT));  // mean(L2)
        const float inv = 1.0f / bw;
        c0 = 4.0f * inv; c1 = 2.0f * inv; c2 = inv; c3 = 0.5f * inv; c4 = 0.25f * inv;
    }
    float local = 0.0f;
#pragma unroll
    for (int mi = 0; mi < 4; ++mi) {
#pragma unroll
        for (int ni = 0; ni < 2; ++ni) {
            const int  colL   = wn * 32 + ni * 16 + l15;
            const float sc    = sqC[colL];
            const bool colSrc = (colBase + colL) < B_ROWS;
#pragma unroll
            for (int r = 0; r < 8; ++r) {
                const int  rowL = wm * 64 + mi * 16 + r + lhalf * 8;   // D layout: m = r + 8*(lane>>4)
                const float sr  = sqR[rowL];
                const float g   = acc[mi][ni][r];
                const float l2  = fmaxf(sr + sc - 2.0f * g, 0.0f);
                if (PHASE == 0) {
                    local += l2;
                } else {
                    const float kern = __expf(-l2 * c0) + __expf(-l2 * c1) + __expf(-l2 * c2)
                                     + __expf(-l2 * c3) + __expf(-l2 * c4);
                    const bool rowSrc = (rowBase + rowL) < B_ROWS;
                    local += (rowSrc == colSrc) ? kern : -kern;
                }
            }
        }
    }
    for (int off = 16; off; off >>= 1) local += __shfl_down(local, off, 32);
    if (lane == 0) red[wave] = local;
    __syncthreads();
    if (tid == 0) {
        float s = 0.0f;
#pragma unroll
        for (int w = 0; w < 8; ++w) s += red[w];
        atomicAdd(&accums[(PHASE == 0) ? 0 : 1], s);
    }
}

__global__ void finalize_kernel(const float* __restrict__ accums, float* __restrict__ out) {
    if (threadIdx.x == 0) {
        const float domain = accums[2] * (1.0f / (2.0f * B_ROWS));
        const float mmd    = accums[1] * (1.0f / ((float)B_ROWS * (float)B_ROWS));
        out[0] = 0.1f * domain + 0.1f * mmd;
        out[1] = domain;
        out[2] = mmd;
    }
}

extern "C" void kernel_launch(void* const* d_in, const int* in_sizes, int n_in,
                              void* d_out, int out_size, void* d_ws, size_t ws_size,
                              hipStream_t stream) {
    const float* src     = (const float*)d_in[0];   // (2048, 512)
    const float* tgt     = (const float*)d_in[1];   // (2048, 512)
    const float* dom_out = (const float*)d_in[2];   // (4096, 1)
    const float* dom_lab = (const float*)d_in[3];   // (4096, 1)
    float* out    = (float*)d_out;                  // 3 floats
    float* wsF    = (float*)d_ws;
    float* accums = wsF;                            // [0]=sumL2 [1]=kernSum [2]=bceSum
    float* sq     = wsF + 16;                       // 4096 floats

    init_kernel<<<1, 32, 0, stream>>>(accums);
    sq_kernel<<<NTOT, 128, 0, stream>>>(src, tgt, sq);
    bce_kernel<<<1, 256, 0, stream>>>(dom_out, dom_lab, accums);

    dim3 grid(NTOT / BT, NTOT / BT);
    mmd_gemm<0><<<grid, 256, 0, stream>>>(src, tgt, sq, accums);   // sum(L2) -> base_bw
    mmd_gemm<1><<<grid, 256, 0, stream>>>(src, tgt, sq, accums);   // signed kernel sum

    finalize_kernel<<<1, 32, 0, stream>>>(accums, out);
}